// HeteroGNNLayer_84782654423761
// MI455X (gfx1250) — compile-verified
//
#include <hip/hip_runtime.h>

typedef __attribute__((ext_vector_type(2))) float v2f;
typedef __attribute__((ext_vector_type(8))) float v8f;

#define EMB    128
#define NREL   3
#define BN_EPS 1e-5f

// ---------------------------------------------------------------------------
// Zero the per-relation aggregation buffer (agg fits in L2: 73.2 MB < 192 MB)
// ---------------------------------------------------------------------------
__global__ void __launch_bounds__(256) zero_f4(float4* __restrict__ p, int n4) {
    int i = blockIdx.x * 256 + threadIdx.x;
    if (i < n4) p[i] = make_float4(0.f, 0.f, 0.f, 0.f);
}

// ---------------------------------------------------------------------------
// K-pair-interleaved transpose:
//   Wt2[r][kp][j][q] = W[r][j][2*kp + q],  kp = 0..63, q = 0..1
// Both K values of a WMMA A/B fragment pair are adjacent -> single b64 load.
// ---------------------------------------------------------------------------
__global__ void __launch_bounds__(256) transpose_W(const float* __restrict__ W,
                                                   float* __restrict__ Wt2) {
    int idx = blockIdx.x * 256 + threadIdx.x;
    if (idx >= NREL * EMB * EMB) return;
    int r  = idx >> 14;             // EMB*EMB = 16384
    int rem = idx & (EMB * EMB - 1);
    int kp = rem >> 8;              // pair index 0..63
    int jq = rem & 255;
    int j  = jq >> 1;
    int q  = jq & 1;
    Wt2[idx] = W[(r << 14) + (j << 7) + 2 * kp + q];
}

// ---------------------------------------------------------------------------
// Per-edge scaled scatter-add:  agg[type][dst] += scale * x[src]
// One wave (32 lanes) per edge; each lane handles 4 floats (float4 gather,
// 4 global fp32 atomics that land in L2-resident lines).
// ---------------------------------------------------------------------------
__global__ void __launch_bounds__(256) edge_scatter(
    const float* __restrict__ x, const int* __restrict__ src,
    const int* __restrict__ dst, const int* __restrict__ etype,
    const float* __restrict__ ew, float* __restrict__ agg,
    int nEdges, int nNodes)
{
    const int lane = threadIdx.x & 31;
    const int e    = blockIdx.x * 8 + (threadIdx.x >> 5);
    if (e >= nEdges) return;

    const int   s = src[e];
    const int   d = dst[e];
    const int   r = etype[e];
    const float scale = (r == 1) ? (ew[e] + 1.0f) * 0.5f : 1.0f;

    const float4 v = *(const float4*)(x + (size_t)s * EMB + lane * 4);
    float* o = agg + ((size_t)r * nNodes + d) * EMB + lane * 4;
    __hip_atomic_fetch_add(o + 0, v.x * scale, __ATOMIC_RELAXED, __HIP_MEMORY_SCOPE_AGENT);
    __hip_atomic_fetch_add(o + 1, v.y * scale, __ATOMIC_RELAXED, __HIP_MEMORY_SCOPE_AGENT);
    __hip_atomic_fetch_add(o + 2, v.z * scale, __ATOMIC_RELAXED, __HIP_MEMORY_SCOPE_AGENT);
    __hip_atomic_fetch_add(o + 3, v.w * scale, __ATOMIC_RELAXED, __HIP_MEMORY_SCOPE_AGENT);
}

// ---------------------------------------------------------------------------
// out = relu( BN( sum_r agg_r @ W_r^T  + x ) )   via V_WMMA_F32_16X16X4_F32.
// One wave per 16-node stripe; wave keeps all 8 N-tiles (16x128 outputs) in
// 8 x v8f accumulators and sweeps K=128 in steps of 4 for each relation.
//
// ISA 32-bit layouts (wave32):
//   A 16x4 : lane%16 = M, reg q + 2*(lane/16) = K
//   B 4x16 : lane%16 = N, reg q + 2*(lane/16) = K
//   C 16x16: reg c -> M = c + 8*(lane/16), N = lane%16
// ---------------------------------------------------------------------------
__global__ void __launch_bounds__(256) gemm_bn_relu(
    const float* __restrict__ agg, const float* __restrict__ Wt2,
    const float* __restrict__ x,
    const float* __restrict__ gamma, const float* __restrict__ beta,
    const float* __restrict__ mean,  const float* __restrict__ var,
    float* __restrict__ out, int nNodes)
{
    const int lane = threadIdx.x & 31;
    const int m    = lane & 15;      // M for A, N for B/C
    const int h    = lane >> 4;      // half-wave select
    const int nodeBase = (blockIdx.x * 8 + (threadIdx.x >> 5)) * 16;
    if (nodeBase >= nNodes) return;

    v8f acc[8] = {};

    for (int r = 0; r < NREL; ++r) {
        const float* A  = agg + ((size_t)r * nNodes + nodeBase + m) * EMB;
        const float* B2 = Wt2 + (size_t)r * EMB * EMB;   // [64][128][2]
        for (int kk = 0; kk < EMB; kk += 4) {
            const int ka = kk + 2 * h;                 // even; pair kp = ka/2
            const v2f a = *(const v2f*)(A + ka);       // A[m][ka], A[m][ka+1]
            const float* bp = B2 + ((size_t)(ka >> 1) * EMB + m) * 2;
#pragma unroll
            for (int t = 0; t < 8; ++t) {
                const v2f b = *(const v2f*)(bp + t * 32);  // (ka, ka+1) for col t*16+m
                acc[t] = __builtin_amdgcn_wmma_f32_16x16x4_f32(
                    false, a, false, b, (short)0, acc[t], false, false);
            }
        }
    }

    // Fused epilogue: residual + eval-mode BatchNorm + ReLU
#pragma unroll
    for (int t = 0; t < 8; ++t) {
        const int   col = t * 16 + m;
        const float sc  = gamma[col] * rsqrtf(var[col] + BN_EPS);
        const float sh  = beta[col] - mean[col] * sc;
#pragma unroll
        for (int c = 0; c < 8; ++c) {
            const int row = nodeBase + c + 8 * h;
            float v = acc[t][c] + x[(size_t)row * EMB + col];
            v = v * sc + sh;
            out[(size_t)row * EMB + col] = fmaxf(v, 0.0f);
        }
    }
}

// ---------------------------------------------------------------------------
extern "C" void kernel_launch(void* const* d_in, const int* in_sizes, int n_in,
                              void* d_out, int out_size, void* d_ws, size_t ws_size,
                              hipStream_t stream)
{
    const float* x     = (const float*)d_in[0];   // (N,128)
    const int*   ei    = (const int*)d_in[1];     // (2,E)
    const int*   et    = (const int*)d_in[2];     // (E,)
    const float* ew    = (const float*)d_in[3];   // (E,)
    const float* W     = (const float*)d_in[4];   // (3,128,128)
    const float* gamma = (const float*)d_in[5];
    const float* beta  = (const float*)d_in[6];
    const float* mean  = (const float*)d_in[7];
    const float* var   = (const float*)d_in[8];
    float*       out   = (float*)d_out;

    const int nNodes = in_sizes[0] / EMB;   // 50000
    const int nEdges = in_sizes[3];         // 600000
    const int* src = ei;
    const int* dst = ei + nEdges;

    // Workspace layout: agg[3][N][128] f32, then Wt2[3][64][128][2] f32
    float* agg = (float*)d_ws;
    float* Wt2 = agg + (size_t)NREL * nNodes * EMB;

    const int n4 = NREL * nNodes * (EMB / 4);
    zero_f4<<<(n4 + 255) / 256, 256, 0, stream>>>((float4*)agg, n4);

    transpose_W<<<(NREL * EMB * EMB + 255) / 256, 256, 0, stream>>>(W, Wt2);

    edge_scatter<<<(nEdges + 7) / 8, 256, 0, stream>>>(
        x, src, dst, et, ew, agg, nEdges, nNodes);

    const int stripes = (nNodes + 15) / 16;
    gemm_bn_relu<<<(stripes + 7) / 8, 256, 0, stream>>>(
        agg, Wt2, x, gamma, beta, mean, var, out, nNodes);
}